// GATConv_11871289606678
// MI455X (gfx1250) — compile-verified
//
#include <hip/hip_runtime.h>
#include <hip/hip_bf16.h>

typedef __attribute__((ext_vector_type(2))) float v2f;
typedef __attribute__((ext_vector_type(4))) float v4f;
typedef __attribute__((ext_vector_type(8))) float v8f;

#define HF 128     // H*F
#define DD 256     // node feature dim
#define DE 8       // edge feature dim
#define NEG_SLOPE 0.2f

// ---------- ordered-int encoding for float atomicMax ----------
__device__ __forceinline__ int enc_f32(float f) {
    int i = __float_as_int(f);
    return i >= 0 ? i : (i ^ 0x7FFFFFFF);
}
__device__ __forceinline__ float dec_f32(int i) {
    int j = i >= 0 ? i : (i ^ 0x7FFFFFFF);
    return __int_as_float(j);
}

// ---------- WMMA f32 16x16x4 step ----------
// Builtin only exists in the device pass; host pass gets a dummy body
// (host never executes __device__ code, it just has to parse it).
__device__ __forceinline__ v8f wmma_f32(v2f a, v2f b, v8f c) {
#if defined(__HIP_DEVICE_COMPILE__)
    return __builtin_amdgcn_wmma_f32_16x16x4_f32(false, a, false, b, (short)0, c,
                                                 false, false);
#else
    (void)a; (void)b;
    return c;
#endif
}

// =====================================================================
// Kernel 0: workspace init (must run every launch — graph replays)
// =====================================================================
__global__ __launch_bounds__(256) void k_init(int* __restrict__ m_enc,
                                              float* __restrict__ z,
                                              float* __restrict__ rst,
                                              int n_seg, int n_rst) {
    int i = blockIdx.x * 256 + threadIdx.x;
    if (i < n_seg) {
        m_enc[i] = (int)0x80000000;  // encoded -inf floor
        z[i] = 0.0f;
    }
    if (i < n_rst) rst[i] = 0.0f;
}

// =====================================================================
// Kernel 1: node projections (WMMA f32) + attention GEMVs
//   h_src = feat @ Wsrc^T            [N,128]
//   h_dst = feat @ Wdst^T + b_dst    [N,128]
//   a_src = feat @ Wattn_src^T       [N,2]
//   a_dst = feat @ Wattn_dst^T       [N,2]
// One workgroup = 16 nodes, 8 waves; feat tile staged in LDS (stride 260).
// =====================================================================
__global__ __launch_bounds__(256) void k_node_proj(
    const float* __restrict__ feat, const float* __restrict__ Wsrc,
    const float* __restrict__ Wdst, const float* __restrict__ bdst,
    const float* __restrict__ Was, const float* __restrict__ Wad,
    float* __restrict__ h_src, float* __restrict__ h_dst,
    float* __restrict__ a_src, float* __restrict__ a_dst, int N_) {
    constexpr int DP = DD + 4;  // 260: 260 % 64 == 4 -> conflict-free rows
    __shared__ float lds[16 * DP];

    const int node0 = blockIdx.x * 16;
    const int t = threadIdx.x;

    for (int idx = t; idx < 16 * DD; idx += 256) {
        int r = idx >> 8, c = idx & (DD - 1);
        lds[r * DP + c] = (node0 + r < N_) ? feat[(size_t)(node0 + r) * DD + c] : 0.0f;
    }
    __syncthreads();

    const int wave  = t >> 5;
    const int lane  = t & 31;
    const int row   = lane & 15;          // A row / B col inside tile
    const int khalf = (lane >> 4) << 1;   // 0 or 2

    const float* brow_s = Wsrc + (size_t)(wave * 16 + row) * DD;
    const float* brow_d = Wdst + (size_t)(wave * 16 + row) * DD;

    v8f acc_s = {}, acc_d = {};
    for (int k = 0; k < DD; k += 4) {
        v2f a  = *(const v2f*)&lds[row * DP + k + khalf];
        v2f bs = *(const v2f*)&brow_s[k + khalf];
        v2f bd = *(const v2f*)&brow_d[k + khalf];
        acc_s = wmma_f32(a, bs, acc_s);
        acc_d = wmma_f32(a, bd, acc_d);
    }

    const int orow0 = (lane >> 4) << 3;  // 0 or 8
    const int col   = wave * 16 + (lane & 15);
    const float bias = bdst[col];
#pragma unroll
    for (int v = 0; v < 8; ++v) {
        int r = node0 + orow0 + v;
        if (r < N_) {
            h_src[(size_t)r * HF + col] = acc_s[v];
            h_dst[(size_t)r * HF + col] = acc_d[v] + bias;
        }
    }

    // attention logit GEMVs: 64 dot products of length 256
    if (t < 64) {
        int which = t >> 4;   // 0:a_src.h0 1:a_src.h1 2:a_dst.h0 3:a_dst.h1
        int r = t & 15;
        const float* w = (which < 2 ? Was : Wad) + (which & 1) * DD;
        float s = 0.0f;
        for (int k = 0; k < DD; ++k) s += lds[r * DP + k] * w[k];
        if (node0 + r < N_) {
            float* outp = (which < 2) ? a_src : a_dst;
            outp[(size_t)(node0 + r) * 2 + (which & 1)] = s;
        }
    }
}

// =====================================================================
// Kernel 2: per-edge logits + LeakyReLU + segment max (ordered-int atomic)
// =====================================================================
__global__ __launch_bounds__(256) void k_edge_logits(
    const float* __restrict__ fe, const int* __restrict__ esrc,
    const int* __restrict__ edst, const float* __restrict__ Wae,
    const float* __restrict__ a_src, const float* __restrict__ a_dst,
    float* __restrict__ ebuf, int* __restrict__ m_enc, int E_) {
    int i = blockIdx.x * 256 + threadIdx.x;
    if (i >= E_) return;
    int s = esrc[i], d = edst[i];
    float f[DE];
#pragma unroll
    for (int j = 0; j < DE; ++j) f[j] = fe[(size_t)i * DE + j];
#pragma unroll
    for (int h = 0; h < 2; ++h) {
        float v = a_src[(size_t)s * 2 + h] + a_dst[(size_t)d * 2 + h];
#pragma unroll
        for (int j = 0; j < DE; ++j) v += f[j] * Wae[h * DE + j];
        v = (v >= 0.0f) ? v : NEG_SLOPE * v;
        ebuf[(size_t)i * 2 + h] = v;
        atomicMax(&m_enc[(size_t)d * 2 + h], enc_f32(v));
    }
}

// =====================================================================
// Kernel 3: exp(e - m) in place + segment sum
// =====================================================================
__global__ __launch_bounds__(256) void k_edge_exp(
    float* __restrict__ ebuf, const int* __restrict__ m_enc,
    const int* __restrict__ edst, float* __restrict__ z, int E_) {
    int i = blockIdx.x * 256 + threadIdx.x;
    if (i >= E_) return;
    int d = edst[i];
#pragma unroll
    for (int h = 0; h < 2; ++h) {
        float m = dec_f32(m_enc[(size_t)d * 2 + h]);
        float ex = expf(ebuf[(size_t)i * 2 + h] - m);
        ebuf[(size_t)i * 2 + h] = ex;
        atomicAdd(&z[(size_t)d * 2 + h], ex);
    }
}

// =====================================================================
// Kernel 4: weighted gather-scatter aggregation (bandwidth-bound hot loop)
// One wave per edge; lane covers 4 consecutive feature columns (float4).
// =====================================================================
__global__ __launch_bounds__(256) void k_aggregate(
    const float* __restrict__ h_src, const float* __restrict__ ebuf,
    const float* __restrict__ z, const int* __restrict__ esrc,
    const int* __restrict__ edst, float* __restrict__ rst, int E_) {
    int e = blockIdx.x * 8 + (threadIdx.x >> 5);
    if (e >= E_) return;
    int lane = threadIdx.x & 31;
    int s = esrc[e], d = edst[e];
    int h = lane >> 4;  // cols 0..63 -> head 0, 64..127 -> head 1
    float coef = ebuf[(size_t)e * 2 + h] / z[(size_t)d * 2 + h];
    v4f hv = *(const v4f*)&h_src[(size_t)s * HF + lane * 4];
    float* rp = &rst[(size_t)d * HF + lane * 4];
    atomicAdd(rp + 0, hv.x * coef);
    atomicAdd(rp + 1, hv.y * coef);
    atomicAdd(rp + 2, hv.z * coef);
    atomicAdd(rp + 3, hv.w * coef);
}

// =====================================================================
// Kernel 5: NGNN GEMM (WMMA f32) + bias + ReLU + residual -> d_out
// =====================================================================
__global__ __launch_bounds__(256) void k_ngnn(
    const float* __restrict__ rst, const float* __restrict__ Wn,
    const float* __restrict__ bn, const float* __restrict__ h_dst,
    float* __restrict__ out, int N_) {
    constexpr int KP = HF + 4;  // 132: 132 % 64 == 4 -> conflict-free rows
    __shared__ float lds[16 * KP];

    const int node0 = blockIdx.x * 16;
    const int t = threadIdx.x;

    for (int idx = t; idx < 16 * HF; idx += 256) {
        int r = idx >> 7, c = idx & (HF - 1);
        lds[r * KP + c] = (node0 + r < N_) ? rst[(size_t)(node0 + r) * HF + c] : 0.0f;
    }
    __syncthreads();

    const int wave  = t >> 5;
    const int lane  = t & 31;
    const int row   = lane & 15;
    const int khalf = (lane >> 4) << 1;

    const float* brow = Wn + (size_t)(wave * 16 + row) * HF;

    v8f acc = {};
    for (int k = 0; k < HF; k += 4) {
        v2f a = *(const v2f*)&lds[row * KP + k + khalf];
        v2f b = *(const v2f*)&brow[k + khalf];
        acc = wmma_f32(a, b, acc);
    }

    const int orow0 = (lane >> 4) << 3;
    const int col   = wave * 16 + (lane & 15);
    const float bias = bn[col];
#pragma unroll
    for (int v = 0; v < 8; ++v) {
        int r = node0 + orow0 + v;
        if (r < N_) {
            float x = acc[v] + bias;
            x = (x > 0.0f) ? x : 0.0f;
            out[(size_t)r * HF + col] = x + h_dst[(size_t)r * HF + col];
        }
    }
}

// =====================================================================
extern "C" void kernel_launch(void* const* d_in, const int* in_sizes, int n_in,
                              void* d_out, int out_size, void* d_ws, size_t ws_size,
                              hipStream_t stream) {
    const float* feat_src = (const float*)d_in[0];   // [N,256]
    const float* feat_edge = (const float*)d_in[1];  // [E,8]
    const int*   edge_src = (const int*)d_in[2];     // [E]
    const int*   edge_dst = (const int*)d_in[3];     // [E]
    const float* W_src = (const float*)d_in[4];      // [128,256]
    const float* W_dst = (const float*)d_in[5];      // [128,256]
    const float* b_dst = (const float*)d_in[6];      // [128]
    const float* W_as  = (const float*)d_in[7];      // [2,256]
    const float* W_ad  = (const float*)d_in[8];      // [2,256]
    const float* W_ae  = (const float*)d_in[9];      // [2,8]
    const float* W_ngnn = (const float*)d_in[10];    // [128,128]
    const float* b_ngnn = (const float*)d_in[11];    // [128]
    float* out = (float*)d_out;

    const int N_ = in_sizes[0] / DD;   // 100000
    const int E_ = in_sizes[2];        // 1600000

    // ---- workspace layout (fp32 elements) ----
    float* ws     = (float*)d_ws;
    float* h_src  = ws;                          // N*128
    float* h_dst  = h_src + (size_t)N_ * HF;     // N*128
    float* rst    = h_dst + (size_t)N_ * HF;     // N*128 (atomic accumulator)
    float* a_src  = rst + (size_t)N_ * HF;       // N*2
    float* a_dst  = a_src + (size_t)N_ * 2;      // N*2
    float* zbuf   = a_dst + (size_t)N_ * 2;      // N*2
    int*   m_enc  = (int*)(zbuf + (size_t)N_ * 2);  // N*2 (ordered-int max)
    float* ebuf   = (float*)(m_enc + (size_t)N_ * 2); // E*2 (logits -> exp)

    const int nb_nodes = (N_ + 15) / 16;
    const int nb_edges = (E_ + 255) / 256;

    // 0) init accumulators (every call: graph replays re-use the buffers)
    {
        int total = N_ * HF;
        k_init<<<(total + 255) / 256, 256, 0, stream>>>(m_enc, zbuf, rst,
                                                        N_ * 2, total);
    }
    // 1) node projections + attention GEMVs (WMMA f32)
    k_node_proj<<<nb_nodes, 256, 0, stream>>>(feat_src, W_src, W_dst, b_dst,
                                              W_as, W_ad, h_src, h_dst, a_src,
                                              a_dst, N_);
    // 2) edge logits + LeakyReLU + segment max
    k_edge_logits<<<nb_edges, 256, 0, stream>>>(feat_edge, edge_src, edge_dst,
                                                W_ae, a_src, a_dst, ebuf, m_enc, E_);
    // 3) exp + segment sum
    k_edge_exp<<<nb_edges, 256, 0, stream>>>(ebuf, m_enc, edge_dst, zbuf, E_);
    // 4) weighted aggregation (gather h_src, atomic scatter into rst)
    k_aggregate<<<(E_ + 7) / 8, 256, 0, stream>>>(h_src, ebuf, zbuf, edge_src,
                                                  edge_dst, rst, E_);
    // 5) NGNN + ReLU + residual (WMMA f32) -> d_out
    k_ngnn<<<nb_nodes, 256, 0, stream>>>(rst, W_ngnn, b_ngnn, h_dst, out, N_);
}